// AttentionFusion3_51127290692312
// MI455X (gfx1250) — compile-verified
//
#include <hip/hip_runtime.h>
#include <math.h>

// ---------------------------------------------------------------------------
// Problem constants (reference: D = 8192, A,B: [D,D] f32, w: [D], b: [1])
// ---------------------------------------------------------------------------
#define DIM 8192
#define STRIPS (DIM / 16)          // 512 row-strips of 16 rows
#define WAVES_PER_BLOCK 8          // 256 threads = 8 wave32
#define CHUNKS_PER_WAVE (DIM / (4 * WAVES_PER_BLOCK))  // 256 K-chunks of 4

typedef __attribute__((ext_vector_type(2))) float v2f;
typedef __attribute__((ext_vector_type(8))) float v8f;

// ---------------------------------------------------------------------------
// Kernel 1: per-strip  Σ_m w[m] * Σ_k A[m,k]*B[m,k]  via V_WMMA_F32_16X16X4_F32
// Block = 16-row strip; wave j takes K-chunks j, j+8, j+16, ... (so the 8
// waves of a block collectively consume each 128B line of the strip once).
// ---------------------------------------------------------------------------
__global__ __launch_bounds__(256) void k_diag_reduce(
    const float* __restrict__ A, const float* __restrict__ B,
    const float* __restrict__ w, float* __restrict__ partials) {
  const int tid  = threadIdx.x;
  const int wave = tid >> 5;
  const int lane = tid & 31;
  const int rowi = lane & 15;        // M (and N) index within strip
  const int half = lane >> 4;        // 0: K={0,1}, 1: K={2,3} of each chunk
  const int strip = blockIdx.x;

  const long row = (long)strip * 16 + rowi;
  const float* __restrict__ arow = A + row * (long)DIM;
  const float* __restrict__ brow = B + row * (long)DIM;

  // Two independent accumulator chains to pipeline the matrix unit.
  v8f acc0 = {};
  v8f acc1 = {};

  const int kofs = 2 * half;
  #pragma unroll 4
  for (int t = 0; t < CHUNKS_PER_WAVE; t += 2) {
    const int k0 = (wave + WAVES_PER_BLOCK * t) * 4 + kofs;
    const int k1 = (wave + WAVES_PER_BLOCK * (t + 1)) * 4 + kofs;
    v2f a0 = *(const v2f*)(arow + k0);
    v2f b0 = *(const v2f*)(brow + k0);
    v2f a1 = *(const v2f*)(arow + k1);
    v2f b1 = *(const v2f*)(brow + k1);
    // D = A(16x4) x Bop(4x16) + C ; Bop[k][n] = B[n,k] -> D[m][n] = Σ_k A[m,k]B[n,k]
    acc0 = __builtin_amdgcn_wmma_f32_16x16x4_f32(
        false, a0, false, b0, (short)0, acc0, false, false);
    acc1 = __builtin_amdgcn_wmma_f32_16x16x4_f32(
        false, a1, false, b1, (short)0, acc1, false, false);
  }

  // Diagonal of the 16x16 accumulator:
  //   VGPR r: lanes 0-15 -> M=r (diag at lane r), lanes 16-31 -> M=r+8
  //   (diag at lane 16+(r+8) = r+24). Dot with w on the fly.
  float part = 0.0f;
  #pragma unroll
  for (int r = 0; r < 8; ++r) {
    const float cr = acc0[r] + acc1[r];
    const float d_lo = __shfl(cr, r, 32);        // diag element M = r
    const float d_hi = __shfl(cr, r + 24, 32);   // diag element M = r + 8
    part += w[strip * 16 + r] * d_lo + w[strip * 16 + r + 8] * d_hi;
  }

  // Block reduction across the 8 waves (each lane of a wave holds same value).
  __shared__ float sp[WAVES_PER_BLOCK];
  if (lane == 0) sp[wave] = part;
  __syncthreads();
  if (tid == 0) {
    float s = 0.0f;
    #pragma unroll
    for (int i = 0; i < WAVES_PER_BLOCK; ++i) s += sp[i];
    partials[strip] = s;
  }
}

// ---------------------------------------------------------------------------
// Kernel 2: reduce 512 strip partials, gate = tanh(sum + b[0])
// ---------------------------------------------------------------------------
__global__ __launch_bounds__(512) void k_gate(
    const float* __restrict__ partials, const float* __restrict__ bias,
    float* __restrict__ gate) {
  __shared__ float red[512];
  const int tid = threadIdx.x;
  red[tid] = partials[tid];
  __syncthreads();
  #pragma unroll
  for (int s = 256; s > 0; s >>= 1) {
    if (tid < s) red[tid] += red[tid + s];
    __syncthreads();
  }
  if (tid == 0) gate[0] = tanhf(red[0] + bias[0]);
}

// ---------------------------------------------------------------------------
// Kernel 3: out = g*A + (1-g)*B, float4-vectorized (bandwidth bound)
// ---------------------------------------------------------------------------
__global__ __launch_bounds__(256) void k_blend(
    const float* __restrict__ A, const float* __restrict__ B,
    const float* __restrict__ gate, float* __restrict__ out) {
  const float g = gate[0];
  const float h = 1.0f - g;
  const size_t idx = ((size_t)blockIdx.x * 256 + threadIdx.x) * 4;
  const float4 a = *(const float4*)(A + idx);
  const float4 b = *(const float4*)(B + idx);
  float4 o;
  o.x = fmaf(g, a.x, h * b.x);
  o.y = fmaf(g, a.y, h * b.y);
  o.z = fmaf(g, a.z, h * b.z);
  o.w = fmaf(g, a.w, h * b.w);
  *(float4*)(out + idx) = o;
}

// ---------------------------------------------------------------------------
// Host-side launcher (graph-capture safe: only kernel launches on `stream`)
// ---------------------------------------------------------------------------
extern "C" void kernel_launch(void* const* d_in, const int* in_sizes, int n_in,
                              void* d_out, int out_size, void* d_ws, size_t ws_size,
                              hipStream_t stream) {
  (void)in_sizes; (void)n_in; (void)out_size; (void)ws_size;
  const float* A = (const float*)d_in[0];
  const float* B = (const float*)d_in[1];
  const float* w = (const float*)d_in[2];
  const float* b = (const float*)d_in[3];
  float* out = (float*)d_out;

  float* ws       = (float*)d_ws;
  float* gate     = ws;        // 1 float
  float* partials = ws + 64;   // 512 floats (256B-aligned offset)

  // Pass 1: gate scalar (every partial written exactly once -> deterministic,
  // no zero-init of workspace needed).
  k_diag_reduce<<<STRIPS, 256, 0, stream>>>(A, B, w, partials);
  k_gate<<<1, 512, 0, stream>>>(partials, b, gate);

  // Pass 2: blend. 8192^2 / (256 threads * 4 floats) = 65536 blocks, no tail.
  const int blend_blocks = (DIM * (DIM / 4)) / 256;
  k_blend<<<blend_blocks, 256, 0, stream>>>(A, B, gate, out);
}